// HaarWaveletTransform2D_6502580486723
// MI455X (gfx1250) — compile-verified
//
#include <hip/hip_runtime.h>

// ---------------------------------------------------------------------------
// HaarWaveletTransform2D  (8,16,512,512) f32 -> (8,64,256,256) f32
//
// Roofline: 268 MB streamed once, ~4 flop/element => AI ~ 0.5 flop/byte.
// Pure HBM-bound (~11.5 us at 23.3 TB/s). WMMA is inapplicable (a 2x2
// butterfly mapped to 16x16 WMMA tiles wastes 75% of each op and adds
// VGPR/LDS shuffles on the bandwidth-critical path).
//
// Design (CDNA5 async-copy pipeline):
//   Phase 1: global_load_async_to_lds_b128 (ASYNCcnt) copies a 16KB tile
//            (8 input rows) per block. Global side: perfectly dense,
//            NT-hinted streaming. LDS side: per-lane scatter splits 16B
//            granules into 4 regions by (row parity, col-granule parity),
//            so phase-2 reads are dense and bank-conflict-free.
//   Phase 2: ds_load_b128 x4 per thread, 2x2 Haar butterfly for 4 output
//            pixels, 4x nontemporal global_store_b128 (one per subband).
//
// NOTE: the LDS destination operand is derived from &lds4[idx] (ptrtoint of
// the shared pointer) so the buffer escapes into the asm; otherwise LLVM
// proves the asm can't write lds4 and folds the phase-2 reads to undef.
// ---------------------------------------------------------------------------

typedef float float4v __attribute__((ext_vector_type(4)));

#define CN 16
#define HN 512
#define WN 512
#define H2 256
#define W2 256
#define SUB_STRIDE   (CN * H2 * W2)       // elements between subband groups
#define TILE_ROWS_IN  8                   // input rows staged per block
#define TILE_GRANULES 1024                // 8 rows * 512 f32 / 4 = 16 KB

__global__ __launch_bounds__(256)
void haar2d_async_kernel(const float* __restrict__ in, float* __restrict__ out) {
    __shared__ float4v lds4[TILE_GRANULES];   // 16 KB staging buffer

    const unsigned bx   = blockIdx.x;
    const unsigned tile = bx & 63u;           // 64 row-tiles per (b,c) plane
    const unsigned c    = (bx >> 6) & 15u;
    const unsigned b    = bx >> 10;
    const unsigned k    = threadIdx.x;        // 0..255

    const float* src = in + (((b * CN + c) * HN) + tile * TILE_ROWS_IN) * WN;

    // ---- Phase 1: async global->LDS with in-flight granule swizzle --------
    // Global granule g (16B): row = g>>7 (2KB rows), colg = g&127.
    // LDS region = (row&1)*2 + (colg&1); index in region = (row>>1)*64 + (colg>>1).
#pragma unroll
    for (unsigned n = 0; n < 4; ++n) {
        unsigned g    = n * 256u + k;
        unsigned row  = g >> 7;
        unsigned colg = g & 127u;
        unsigned p    = row & 1u;             // input-row parity
        unsigned r    = row >> 1;             // output row within tile (0..3)
        unsigned q    = colg & 1u;            // granule parity within pixel pair
        unsigned jv   = colg >> 1;            // output col group (0..63)
        unsigned gran = (((p << 1) | q) << 8) | (r << 6) | jv;

        // LDS byte address derived from the shared pointer itself:
        // low 32 bits of the flat address == LDS byte address (ISA 10.2),
        // and the ptrtoint use makes lds4 escape into the asm.
        unsigned lds_off = (unsigned)(unsigned long long)(uintptr_t)&lds4[gran];
        const float* gp = src + g * 4u;       // dense 16B granule stream
        asm volatile("global_load_async_to_lds_b128 %0, %1, off th:TH_LOAD_NT"
                     :: "v"(lds_off), "v"(gp) : "memory");
    }
    asm volatile("s_wait_asynccnt 0" ::: "memory");   // own copies complete
    __syncthreads();                                  // cross-wave visibility

    // ---- Phase 2: dense, conflict-free LDS reads + butterfly --------------
    // Thread k: r = k>>6, jv = k&63 ; granule index within each region == k.
    float4v a0 = lds4[         k];    // row 2r,   cols 8jv..8jv+3
    float4v a1 = lds4[256u  +  k];    // row 2r,   cols 8jv+4..8jv+7
    float4v b0 = lds4[512u  +  k];    // row 2r+1, cols 8jv..8jv+3
    float4v b1 = lds4[768u  +  k];    // row 2r+1, cols 8jv+4..8jv+7

    float4v s0, d0, s1, d1;
    s0.x = a0.x + a0.y;  d0.x = a0.x - a0.y;
    s0.y = a0.z + a0.w;  d0.y = a0.z - a0.w;
    s0.z = a1.x + a1.y;  d0.z = a1.x - a1.y;
    s0.w = a1.z + a1.w;  d0.w = a1.z - a1.w;
    s1.x = b0.x + b0.y;  d1.x = b0.x - b0.y;
    s1.y = b0.z + b0.w;  d1.y = b0.z - b0.w;
    s1.z = b1.x + b1.y;  d1.z = b1.x - b1.y;
    s1.w = b1.z + b1.w;  d1.w = b1.z - b1.w;

    float4v ll = (s0 + s1) * 0.5f;   // low_low
    float4v lh = (s0 - s1) * 0.5f;   // low_high
    float4v hl = (d0 + d1) * 0.5f;   // high_low
    float4v hh = (d0 - d1) * 0.5f;   // high_high

    const unsigned r  = k >> 6;
    const unsigned jv = k & 63u;
    const unsigned i  = tile * 4u + r;                 // output row
    const unsigned outBase = ((b * 4u * CN + c) * H2 + i) * W2 + 4u * jv;

    __builtin_nontemporal_store(ll, reinterpret_cast<float4v*>(out + outBase));
    __builtin_nontemporal_store(lh, reinterpret_cast<float4v*>(out + outBase + 1u * SUB_STRIDE));
    __builtin_nontemporal_store(hl, reinterpret_cast<float4v*>(out + outBase + 2u * SUB_STRIDE));
    __builtin_nontemporal_store(hh, reinterpret_cast<float4v*>(out + outBase + 3u * SUB_STRIDE));
}

// ---------------------------------------------------------------------------
// Retained compile-probe (never launched): validated 6-arg TDM builtin on
// this toolchain (clang-23 / therock headers). TDM is not used in the live
// kernel because its descriptor walk cannot express the parity-split granule
// scatter that makes the LDS reads conflict-free; the per-lane async copy can.
// ---------------------------------------------------------------------------
typedef unsigned int uint32x4 __attribute__((ext_vector_type(4)));
typedef int          int32x8  __attribute__((ext_vector_type(8)));
typedef int          int32x4  __attribute__((ext_vector_type(4)));
__global__ void cdna5_probe_tdm() {
    uint32x4 g0 = {0u, 0u, 0u, 0u};
    int32x8  g1 = {0, 0, 0, 0, 0, 0, 0, 0};
    int32x4  g2 = {0, 0, 0, 0};
    int32x4  g3 = {0, 0, 0, 0};
    int32x8  g4 = {0, 0, 0, 0, 0, 0, 0, 0};
    __builtin_amdgcn_tensor_load_to_lds(g0, g1, g2, g3, g4, 0);
    __builtin_amdgcn_s_wait_tensorcnt(0);
}

// ---------------------------------------------------------------------------

extern "C" void kernel_launch(void* const* d_in, const int* in_sizes, int n_in,
                              void* d_out, int out_size, void* d_ws, size_t ws_size,
                              hipStream_t stream) {
    const float* x = (const float*)d_in[0];
    float* out = (float*)d_out;

    // blocks = planes * row-tiles = (n / (512*512)) * 64 = 8192
    unsigned n      = (unsigned)in_sizes[0];     // 33,554,432
    unsigned planes = n / (unsigned)(HN * WN);   // 128
    unsigned blocks = planes * 64u;              // 8192

    haar2d_async_kernel<<<blocks, 256, 0, stream>>>(x, out);
}